// MyCrossAttention_28544352650054
// MI455X (gfx1250) — compile-verified
//
#include <hip/hip_runtime.h>

// ---------------------------------------------------------------------------
// Prompt-to-Prompt cross attention for MI455X (gfx1250), fp32 WMMA path.
//
// Shapes: B=8, T=4096, C=320, H=8, d=40, S=77, D=768.
// All matmul stages use V_WMMA_F32_16X16X4_F32 (wave32, 16x16 C/D tile,
// A = 16x4 fp32 frag (v2f / lane), B = 4x16 fp32 frag (v2f / lane)).
//
// Fragment layout assumptions (CDNA5 ISA 7.12.2):
//   A 16x4 : lanes 0-15 hold row M=lane, K = k0,k0+1 ; lanes 16-31 K = k0+2,k0+3
//   B 4x16 : VGPR0 = row k0 (lanes 0-15) / row k0+2 (lanes 16-31), VGPR1 = +1
//   C/D    : VGPR i -> row i (lanes 0-15) / row i+8 (lanes 16-31), col = lane&15
// ---------------------------------------------------------------------------

typedef __attribute__((ext_vector_type(2))) float v2f;
typedef __attribute__((ext_vector_type(4))) float v4f;
typedef __attribute__((ext_vector_type(8))) float v8f;

#define B_ALL   8
#define B_BASE  4      // cond base batch (attn_base source)
#define T_DIM   4096
#define C_DIM   320
#define S_DIM   77
#define SP      80     // S padded to multiple of 16 (zeros in pad)
#define D_ENC   768
#define HD      40     // head dim
#define H_NUM   8

__device__ __forceinline__ v8f wmma4(v2f a, v2f b, v8f c) {
  // D = A(16x4,f32) * B(4x16,f32) + C(16x16,f32)
  return __builtin_amdgcn_wmma_f32_16x16x4_f32(
      /*neg_a=*/false, a, /*neg_b=*/false, b,
      /*c_mod=*/(short)0, c, /*reuse_a=*/false, /*reuse_b=*/false);
}

// ---------------------------------------------------------------------------
// Row-major GEMM: C[M,N] = A[M,K] * B[K,N] (+ bias[N]), lda=K, ldb=ldc=N.
// 128 threads = 4 waves per block; block computes a 64x64 C tile (wave w owns
// rows rowW..rowW+15, all 64 cols -> 4 accumulators -> 4x A-frag reuse).
// B is double-buffered in LDS, staged TRANSPOSED (Bs[c*18+k]) so the fragment
// read is one aligned ds_load_b64; stride 18 keeps banks spread. A rows are
// clamped (not predicated): a WMMA row only feeds its own output row, and
// out-of-range rows are never stored, so no EXEC toggling in the hot loop.
// Requires: N multiple of 64, K multiple of 16 (true at every call site).
// ---------------------------------------------------------------------------
#define GK 16   // K-block

__global__ __launch_bounds__(128)
void gemm_wmma_f32(const float* __restrict__ A, const float* __restrict__ Bm,
                   const float* __restrict__ bias, float* __restrict__ C,
                   int M, int N, int K) {
  __shared__ float Bs[2][64 * 18];
  const int tid  = threadIdx.x;
  const int wave = tid >> 5;
  const int lane = tid & 31;
  const int half = lane >> 4;
  const int l16  = lane & 15;
  const int col0 = blockIdx.x * 64;
  const int rowW = blockIdx.y * 64 + wave * 16;

  // clamped A row pointer (loads always in-bounds; stores guarded later)
  int arow = rowW + l16;
  if (arow > M - 1) arow = M - 1;
  const float* arowp = A + (size_t)arow * K;

  // cooperative B staging: thread loads 8 consecutive floats of one B row
  const int srow = tid >> 3;        // 0..15  (k within block)
  const int scol = (tid & 7) * 8;   // 0..56  (col within 64-wide tile)
  const float* bstage = Bm + (size_t)srow * N + col0 + scol;

  const v8f zero = {0.f,0.f,0.f,0.f,0.f,0.f,0.f,0.f};
  v8f acc[4];
  #pragma unroll
  for (int j = 0; j < 4; ++j) acc[j] = zero;

  const int nkb = K / GK;

  // prologue: stage block 0
  {
    v4f r0 = *(const v4f*)(bstage);
    v4f r1 = *(const v4f*)(bstage + 4);
    #pragma unroll
    for (int e = 0; e < 4; ++e) Bs[0][(scol + e) * 18 + srow] = r0[e];
    #pragma unroll
    for (int e = 0; e < 4; ++e) Bs[0][(scol + 4 + e) * 18 + srow] = r1[e];
  }
  __syncthreads();

  for (int kb = 0; kb < nkb; ++kb) {
    const int cur = kb & 1;
    const bool have_next = (kb + 1) < nkb;

    // issue next block's global loads early (hidden behind current compute)
    v4f r0, r1;
    if (have_next) {
      const float* bnxt = bstage + (size_t)(kb + 1) * GK * N;
      r0 = *(const v4f*)(bnxt);
      r1 = *(const v4f*)(bnxt + 4);
      if (kb + 2 < nkb)
        __builtin_prefetch(bstage + (size_t)(kb + 2) * GK * N, 0, 1);
    }

    // compute: 4 k-steps x 4 col-tiles = 16 WMMA per K-block
    const float* ap = arowp + kb * GK;
    #pragma unroll
    for (int kk = 0; kk < GK; kk += 4) {
      const int ka = kk + 2 * half;
      v2f a = *(const v2f*)(ap + ka);
      #pragma unroll
      for (int j = 0; j < 4; ++j) {
        v2f b = *(const v2f*)(&Bs[cur][(16 * j + l16) * 18 + ka]);
        acc[j] = wmma4(a, b, acc[j]);
      }
    }

    // store next block into the other buffer (its last readers finished
    // before the previous barrier), then one barrier per K-block
    if (have_next) {
      const int nxt = cur ^ 1;
      #pragma unroll
      for (int e = 0; e < 4; ++e) Bs[nxt][(scol + e) * 18 + srow] = r0[e];
      #pragma unroll
      for (int e = 0; e < 4; ++e) Bs[nxt][(scol + 4 + e) * 18 + srow] = r1[e];
    }
    __syncthreads();
  }

  const float bv = bias ? bias[col0 + 16 * 0 + l16] : 0.f; // per-tile below
  #pragma unroll
  for (int j = 0; j < 4; ++j) {
    const int c = col0 + 16 * j + l16;
    const float bj = bias ? bias[c] : 0.f;
    #pragma unroll
    for (int i = 0; i < 8; ++i) {
      const int r = rowW + i + 8 * half;
      if (r < M) C[(size_t)r * N + c] = acc[j][i] + bj;
    }
  }
  (void)bv;
}

// ---------------------------------------------------------------------------
// P(16xSP tile, in LDS) @ V(b)[77x40 head slice] -> AO[b][t0..t0+15][h*40..+40)
// K loop over SP=80 (pad rows of P are zero). 3 N-tiles cover 40 cols (pad 48).
// ---------------------------------------------------------------------------
__device__ __forceinline__
void pv_and_store(const float* Plds, const float* __restrict__ Vp,
                  float* __restrict__ AO, int b, int h, int t0,
                  int half, int l16) {
  const v8f zero = {0.f,0.f,0.f,0.f,0.f,0.f,0.f,0.f};
  v8f o[3];
  #pragma unroll
  for (int j = 0; j < 3; ++j) o[j] = zero;

  const float* vbase = Vp + (size_t)(b * S_DIM) * C_DIM + h * HD;

  for (int k0 = 0; k0 < SP; k0 += 4) {
    const int ka = k0 + 2 * half;
    v2f a = *(const v2f*)(Plds + l16 * SP + ka);   // 8B-aligned (SP even, ka even)
    #pragma unroll
    for (int j = 0; j < 3; ++j) {
      const int c = 16 * j + l16;
      v2f bf; bf.x = 0.f; bf.y = 0.f;
      if (c < HD) {
        if (ka     < S_DIM) bf.x = vbase[(size_t) ka      * C_DIM + c];
        if (ka + 1 < S_DIM) bf.y = vbase[(size_t)(ka + 1) * C_DIM + c];
      }
      o[j] = wmma4(a, bf, o[j]);
    }
  }

  #pragma unroll
  for (int j = 0; j < 3; ++j) {
    const int c = 16 * j + l16;
    if (c < HD) {
      #pragma unroll
      for (int i = 0; i < 8; ++i) {
        const int t = t0 + i + 8 * half;
        AO[(size_t)(b * T_DIM + t) * C_DIM + h * HD + c] = o[j][i];
      }
    }
  }
}

// ---------------------------------------------------------------------------
// Fused scores + softmax + P@V for batches 0..4. One wave per (b, h, 16 rows).
// b==4 additionally spills normalized probs (attn_base) to AB[H][T][SP].
// ---------------------------------------------------------------------------
__global__ __launch_bounds__(32)
void attn_kernel(const float* __restrict__ Q,   // [5*T, 320]
                 const float* __restrict__ Kp,  // [8*77, 320]
                 const float* __restrict__ Vp,  // [8*77, 320]
                 float* __restrict__ AB,        // [H][T][SP]
                 float* __restrict__ AO)        // [8*T, 320]
{
  __shared__ float Plds[16 * SP];
  const int lane = threadIdx.x;
  const int half = lane >> 4;
  const int l16  = lane & 15;
  const int t0 = blockIdx.x << 4;
  const int h  = blockIdx.y;
  const int b  = blockIdx.z;                 // 0..4
  const float scale = 0.15811388300841898f;  // 1/sqrt(40)
  const float NEG_INF = -__builtin_inff();

  // ---- scores: Q(16x40) @ K^T(40x77->80), 5 N-tiles x 10 K-steps = 50 WMMA
  const v8f zero = {0.f,0.f,0.f,0.f,0.f,0.f,0.f,0.f};
  v8f sc[5];
  #pragma unroll
  for (int j = 0; j < 5; ++j) sc[j] = zero;

  const float* qbase = Q  + (size_t)(b * T_DIM + t0 + l16) * C_DIM + h * HD;
  const float* kbase = Kp + (size_t)(b * S_DIM) * C_DIM + h * HD;

  for (int k0 = 0; k0 < HD; k0 += 4) {
    const int ka = k0 + 2 * half;
    v2f a = *(const v2f*)(qbase + ka);
    #pragma unroll
    for (int j = 0; j < 5; ++j) {
      const int s = 16 * j + l16;
      v2f bf;
      if (s < S_DIM) bf = *(const v2f*)(kbase + (size_t)s * C_DIM + ka);
      else { bf.x = 0.f; bf.y = 0.f; }
      sc[j] = wmma4(a, bf, sc[j]);
    }
  }

  // ---- scale + mask + row softmax (rows live across 16-lane halves)
  float mx[8];
  #pragma unroll
  for (int i = 0; i < 8; ++i) mx[i] = NEG_INF;
  #pragma unroll
  for (int j = 0; j < 5; ++j) {
    const bool ok = (16 * j + l16) < S_DIM;
    #pragma unroll
    for (int i = 0; i < 8; ++i) {
      const float x = ok ? sc[j][i] * scale : NEG_INF;
      sc[j][i] = x;
      mx[i] = fmaxf(mx[i], x);
    }
  }
  #pragma unroll
  for (int m = 1; m < 16; m <<= 1) {
    #pragma unroll
    for (int i = 0; i < 8; ++i) mx[i] = fmaxf(mx[i], __shfl_xor(mx[i], m, 32));
  }
  float sm[8];
  #pragma unroll
  for (int i = 0; i < 8; ++i) sm[i] = 0.f;
  #pragma unroll
  for (int j = 0; j < 5; ++j) {
    #pragma unroll
    for (int i = 0; i < 8; ++i) {
      const float e = __expf(sc[j][i] - mx[i]);  // exp(-inf)=0 handles the pad
      sc[j][i] = e;
      sm[i] += e;
    }
  }
  #pragma unroll
  for (int m = 1; m < 16; m <<= 1) {
    #pragma unroll
    for (int i = 0; i < 8; ++i) sm[i] += __shfl_xor(sm[i], m, 32);
  }
  float rs[8];
  #pragma unroll
  for (int i = 0; i < 8; ++i) rs[i] = 1.f / sm[i];

  // ---- stage normalized P in LDS (and spill attn_base for b==4)
  #pragma unroll
  for (int j = 0; j < 5; ++j) {
    #pragma unroll
    for (int i = 0; i < 8; ++i) {
      const float pv = sc[j][i] * rs[i];
      const int row = i + 8 * half;
      Plds[row * SP + 16 * j + l16] = pv;
      if (b == B_BASE)
        AB[(size_t)(h * T_DIM + t0 + row) * SP + 16 * j + l16] = pv;
    }
  }
  __syncthreads();

  pv_and_store(Plds, Vp, AO, b, h, t0, half, l16);
}

// ---------------------------------------------------------------------------
// Batches 5..7: P_p = attn_base @ mapper[p]  (16x80 tile, K=80 w/ zero pad),
// then same P@V path. Their own Q/K are dead in the reference — skipped.
// ---------------------------------------------------------------------------
__global__ __launch_bounds__(32)
void edit_kernel(const float* __restrict__ AB,    // [H][T][SP]
                 const float* __restrict__ Map,   // [3][77][77]
                 const float* __restrict__ Vp,    // [8*77, 320]
                 float* __restrict__ AO)          // [8*T, 320]
{
  __shared__ float Plds[16 * SP];
  const int lane = threadIdx.x;
  const int half = lane >> 4;
  const int l16  = lane & 15;
  const int t0 = blockIdx.x << 4;
  const int h  = blockIdx.y;
  const int p  = blockIdx.z;        // 0..2
  const int b  = B_BASE + 1 + p;    // 5..7

  const v8f zero = {0.f,0.f,0.f,0.f,0.f,0.f,0.f,0.f};
  v8f acc[5];
  #pragma unroll
  for (int j = 0; j < 5; ++j) acc[j] = zero;

  const float* abase = AB  + (size_t)(h * T_DIM + t0 + l16) * SP;
  const float* mbase = Map + (size_t)p * S_DIM * S_DIM;

  for (int k0 = 0; k0 < SP; k0 += 4) {
    const int ka = k0 + 2 * half;
    v2f a = *(const v2f*)(abase + ka);      // pad cols of AB are stored zeros
    #pragma unroll
    for (int j = 0; j < 5; ++j) {
      const int n = 16 * j + l16;
      v2f bf; bf.x = 0.f; bf.y = 0.f;
      if (n < S_DIM) {
        if (ka     < S_DIM) bf.x = mbase[(size_t) ka      * S_DIM + n];
        if (ka + 1 < S_DIM) bf.y = mbase[(size_t)(ka + 1) * S_DIM + n];
      }
      acc[j] = wmma4(a, bf, acc[j]);
    }
  }

  #pragma unroll
  for (int j = 0; j < 5; ++j) {
    const bool ok = (16 * j + l16) < S_DIM;
    #pragma unroll
    for (int i = 0; i < 8; ++i) {
      const int row = i + 8 * half;
      Plds[row * SP + 16 * j + l16] = ok ? acc[j][i] : 0.f;
    }
  }
  __syncthreads();

  pv_and_store(Plds, Vp, AO, b, h, t0, half, l16);
}

// ---------------------------------------------------------------------------
// Launch: Q/K/V projections -> fused attention (b0..4) -> edit (b5..7) -> Wo.
// Workspace layout (floats): Q[20480*320] | Kp[616*320] | Vp[616*320]
//                            | AB[8*4096*80] | AO[32768*320]  (~80 MB)
// ---------------------------------------------------------------------------
extern "C" void kernel_launch(void* const* d_in, const int* in_sizes, int n_in,
                              void* d_out, int out_size, void* d_ws, size_t ws_size,
                              hipStream_t stream) {
  const float* X   = (const float*)d_in[0];  // hidden_states [8,4096,320]
  const float* E   = (const float*)d_in[1];  // encoder_hidden_states [8,77,768]
  const float* Map = (const float*)d_in[2];  // mapper [3,77,77]
  const float* Wq  = (const float*)d_in[3];
  const float* Wk  = (const float*)d_in[4];
  const float* Wv  = (const float*)d_in[5];
  const float* Wo  = (const float*)d_in[6];
  const float* bo  = (const float*)d_in[7];
  float* out = (float*)d_out;

  float* ws = (float*)d_ws;
  float* Q  = ws;                                 // 5*4096 x 320
  float* Kp = Q  + (size_t)5 * T_DIM * C_DIM;     // 616 x 320
  float* Vp = Kp + (size_t)B_ALL * S_DIM * C_DIM;
  float* AB = Vp + (size_t)B_ALL * S_DIM * C_DIM; // 8 x 4096 x 80
  float* AO = AB + (size_t)H_NUM * T_DIM * SP;    // 32768 x 320

  const dim3 gblk(128);
  const dim3 ablk(32);
  const int MR = B_ALL * S_DIM;                   // 616 rows for K/V proj

  // Q projection (only batches 0..4 are live), K/V projections
  gemm_wmma_f32<<<dim3(C_DIM / 64, (5 * T_DIM) / 64), gblk, 0, stream>>>(
      X, Wq, nullptr, Q, 5 * T_DIM, C_DIM, C_DIM);
  gemm_wmma_f32<<<dim3(C_DIM / 64, (MR + 63) / 64), gblk, 0, stream>>>(
      E, Wk, nullptr, Kp, MR, C_DIM, D_ENC);
  gemm_wmma_f32<<<dim3(C_DIM / 64, (MR + 63) / 64), gblk, 0, stream>>>(
      E, Wv, nullptr, Vp, MR, C_DIM, D_ENC);

  // Fused attention for batches 0..4 (writes AB for b==4)
  attn_kernel<<<dim3(T_DIM / 16, H_NUM, 5), ablk, 0, stream>>>(Q, Kp, Vp, AB, AO);

  // Edited attention for batches 5..7
  edit_kernel<<<dim3(T_DIM / 16, H_NUM, 3), ablk, 0, stream>>>(AB, Map, Vp, AO);

  // Output projection with bias
  gemm_wmma_f32<<<dim3(C_DIM / 64, (B_ALL * T_DIM) / 64), gblk, 0, stream>>>(
      AO, Wo, bo, out, B_ALL * T_DIM, C_DIM, C_DIM);
}